// IrrepsIndexedLinear_39161511805249
// MI455X (gfx1250) — compile-verified
//
#include <hip/hip_runtime.h>

typedef __attribute__((ext_vector_type(2))) float v2f;
typedef __attribute__((ext_vector_type(8))) float v8f;

// Explicit global-address-space pointers so x/w/y traffic lowers to
// global_load/global_store (not flat_*, which would tax DScnt + the LDS pipe).
typedef __attribute__((address_space(1))) const float* gcfp;
typedef __attribute__((address_space(1))) float*       gfp;
typedef __attribute__((address_space(1))) const int*   gcip;

#define MUL_C    128
#define G_GROUPS 32
#define N_ROWS   2048

// One 16x16 output tile:  D(16x16) += A(16x4) * B(4x16), K looped 128/4 = 32.
// D is the irrep dimension (1/3/5), compile-time so all div/mod folds.
template<int D>
__device__ __forceinline__ void compute_tile(
    gcfp xbase, gfp ob_ir, const v2f* __restrict__ lb,
    int mtile, int startg, int cnt, int lane, int op, int khalf, int nt)
{
    const int mrows = cnt * D;

    // A row r=(n,i) for this lane; clamp OOB rows to a safe in-bounds address
    // (their D rows are never stored) so EXEC stays all-1s through the WMMAs.
    const int r  = mtile * 16 + op;
    const int rc = (r < mrows) ? r : (mrows - 1);
    const int n  = startg + rc / D;
    const int i  = rc % D;
    gcfp pa = xbase + (size_t)n * (MUL_C * D) + i + khalf * D;

    v8f acc = {0.f, 0.f, 0.f, 0.f, 0.f, 0.f, 0.f, 0.f};

    #pragma unroll 8
    for (int kk = 0; kk < 32; ++kk) {
        v2f a;
        a.x = pa[0];            // A[m][k]    (D==1: merges into one b64 load)
        a.y = pa[D];            // A[m][k+1]
        pa += 4 * D;
        v2f b = lb[(kk * 2 + (lane >> 4)) * 16 + op];   // B[k..k+1][o] from LDS
        acc = __builtin_amdgcn_wmma_f32_16x16x4_f32(
            false, a, false, b, (short)0, acc, false, false);
    }

    // D layout: lane holds N=op; VGPR v holds M = v + 8*(lane>>4).
    // One division, then incremental (n,i)/address updates per element.
    const int mbase = mtile * 16 + ((lane >> 4) << 3);
    const int o     = nt * 16 + op;
    int nn = startg + mbase / D;
    int ii = mbase % D;
    size_t idx = ((size_t)nn * MUL_C + o) * D + ii;

    #pragma unroll
    for (int v = 0; v < 8; ++v) {
        const int rr = mbase + v;
        if (rr < mrows) ob_ir[idx] = acc[v] * 0.015625f;   // alpha = 1/64
        const bool wrap = (ii == D - 1);
        idx += wrap ? (size_t)(MUL_C * D - (D - 1)) : (size_t)1;
        ii   = wrap ? 0 : ii + 1;
    }
}

__global__ __launch_bounds__(256) void irreps_indexed_linear_wmma(
    const float* __restrict__ x0,
    const float* __restrict__ x1,
    const float* __restrict__ x2,
    const float* __restrict__ w,
    const int*   __restrict__ repeats,
    float* __restrict__ out)
{
    // [ir][(k>>1)*16 + o] holds the (k, k+1) weight pair for output col o.
    __shared__ v2f wlds[3][1024];

    const int g    = blockIdx.x >> 3;   // weight group
    const int nt   = blockIdx.x & 7;    // output-column tile (16 cols)
    const int tid  = threadIdx.x;
    const int lane = tid & 31;          // wave32
    const int wave = tid >> 5;          // 8 waves / block

    // Uniform scalar scan for this group's row range.
    gcip rep = (gcip)repeats;
    int startg = 0;
    for (int i = 0; i < G_GROUPS; ++i) {
        int c = rep[i];
        if (i < g) startg += c;
    }
    const int cnt = rep[g];

    // Stage W[128 x 16] per irrep into LDS, pair-interleaved for b64 B loads.
    {
        float* lf = (float*)&wlds[0][0];
        gcfp wg = (gcfp)w + (size_t)g * (3 * MUL_C * MUL_C);
        for (int ir = 0; ir < 3; ++ir) {
            gcfp wi = wg + ir * MUL_C * MUL_C + nt * 16;
            float* li = lf + ir * 2048;
            for (int t = tid; t < 2048; t += 256) {
                const int k = t >> 4;       // mul_in index (K)
                const int o = t & 15;       // local output col
                const float v = wi[k * MUL_C + o];
                const int f = (((k >> 1) * 16 + o) << 1) | (k & 1);
                li[f] = v;
            }
        }
    }
    __syncthreads();

    if (cnt == 0) return;

    const int mt0 = (cnt * 1 + 15) >> 4;
    const int mt1 = (cnt * 3 + 15) >> 4;
    const int mt2 = (cnt * 5 + 15) >> 4;
    const int total = mt0 + mt1 + mt2;

    const int op    = lane & 15;            // A row (M) on load side; D col (N) on store side
    const int khalf = (lane >> 4) << 1;     // 0 or 2: this lane's K sub-pair

    gfp out0 = (gfp)out;
    gfp out1 = (gfp)(out + (size_t)N_ROWS * MUL_C);
    gfp out2 = (gfp)(out + (size_t)N_ROWS * MUL_C * 4);

    // t, mt0, mt1, total are wave-uniform -> scalar branches, no divergence.
    for (int t = wave; t < total; t += 8) {
        if (t < mt0) {
            compute_tile<1>((gcfp)x0, out0, &wlds[0][0], t,
                            startg, cnt, lane, op, khalf, nt);
        } else if (t < mt0 + mt1) {
            compute_tile<3>((gcfp)x1, out1, &wlds[1][0], t - mt0,
                            startg, cnt, lane, op, khalf, nt);
        } else {
            compute_tile<5>((gcfp)x2, out2, &wlds[2][0], t - mt0 - mt1,
                            startg, cnt, lane, op, khalf, nt);
        }
    }
}

extern "C" void kernel_launch(void* const* d_in, const int* in_sizes, int n_in,
                              void* d_out, int out_size, void* d_ws, size_t ws_size,
                              hipStream_t stream) {
    (void)in_sizes; (void)n_in; (void)out_size; (void)d_ws; (void)ws_size;
    const float* x0      = (const float*)d_in[0];
    const float* x1      = (const float*)d_in[1];
    const float* x2      = (const float*)d_in[2];
    const float* w       = (const float*)d_in[3];
    const int*   repeats = (const int*)d_in[4];
    float*       out     = (float*)d_out;

    dim3 grid(G_GROUPS * 8);   // 32 groups x 8 output-column tiles
    dim3 block(256);           // 8 wave32 waves
    irreps_indexed_linear_wmma<<<grid, block, 0, stream>>>(x0, x1, x2, w, repeats, out);
}